// ALiBi_29618094473876
// MI455X (gfx1250) — compile-verified
//
#include <hip/hip_runtime.h>

// ALiBi bias subtraction: out = scores - slope(h) * (i - j)
//   scores: [B=2, H=16, S=2048, S=2048] f32.  `offset` cancels exactly.
//
// HBM-bound: 1 GiB traffic -> ~46us floor at 23.3 TB/s. Streaming (no reuse,
// 1 GiB >> 192 MB L2) -> non-temporal loads/stores. The bias is rank-2 over a
// 16x16 tile, so a single V_WMMA_F32_16X16X4_F32 computes D = A x B + C with
// C = scores tile:
//   A col set u: -0.5*slope*(i0+m), col set v: 0.5
//   B row set u:  1.0,              row set v: slope*(j0+n)
//   => A x B = -slope*(i0+m) + slope*(j0+n) = -slope*(i-j)
// Values are duplicated across lane halves so the result is invariant to the
// exact (vgpr, lane-half) -> K slotting, provided A and B share the mapping.

typedef __attribute__((ext_vector_type(2))) float v2f;
typedef __attribute__((ext_vector_type(8))) float v8f;

#define B_DIM 2
#define H_DIM 16
#define S_DIM 2048
#define TILES_J_PER_WAVE 8   // each wave handles a 16 x 128 strip

__global__ __launch_bounds__(256) void alibi_wmma_kernel(
    const float* __restrict__ scores, float* __restrict__ out)
{
    const int lane = threadIdx.x & 31;
    const int wave = threadIdx.x >> 5;
    const int n    = lane & 15;   // N (and M) index within tile
    const int hihalf = lane >> 4; // 0 for lanes 0-15, 1 for lanes 16-31

    // Global wave id -> (bh, i-tile, j-strip)
    const int w      = blockIdx.x * 8 + wave;
    const int jstrip = w & 15;           // 16 strips of 128 columns
    const int itile  = (w >> 4) & 127;   // 128 row tiles
    const int bh     = w >> 11;          // b*H + h  (0..31)
    const int h      = bh & (H_DIM - 1);

    // slope = 2^(-8*(h+1)/H) = 2^(-(h+1)/2): exact pow2 times RN(1/sqrt(2))
    const int e = h + 1;
    float slope = __builtin_ldexpf(1.0f, -(e >> 1));
    if (e & 1) slope *= 0x1.6a09e6p-1f;  // 2^-0.5 correctly rounded to f32

    const int i0 = itile * 16;
    const int j0 = jstrip * (16 * TILES_J_PER_WAVE);

    // A matrix (16x4 f32 -> 2 VGPRs), M = lane&15 in both halves.
    v2f a;
    a.x = -0.5f * slope * (float)(i0 + n);
    a.y = 0.5f;

    const size_t base = (size_t)bh * S_DIM * S_DIM;
    const size_t row0 = (size_t)(i0 + hihalf * 8) * S_DIM;
    const float* src = scores + base + row0 + n;
    float*       dst = out    + base + row0 + n;

    for (int t = 0; t < TILES_J_PER_WAVE; ++t) {
        const int j = j0 + t * 16;

        // B matrix (4x16 f32 -> 2 VGPRs), N = lane&15 in both halves.
        v2f b;
        b.x = 1.0f;
        b.y = slope * (float)(j + n);

        // C = scores tile per the 16x16 f32 C/D layout:
        // VGPR k holds row (k + 8*(lane>=16)), column lane&15.
        v8f c;
#pragma unroll
        for (int k = 0; k < 8; ++k)
            c[k] = __builtin_nontemporal_load(src + (size_t)k * S_DIM + j);

        // D = A*B + C = scores - slope*(i-j)
        v8f d = __builtin_amdgcn_wmma_f32_16x16x4_f32(
            /*neg_a=*/false, a, /*neg_b=*/false, b,
            /*c_mod=*/(short)0, c, /*reuse_a=*/false, /*reuse_b=*/false);

#pragma unroll
        for (int k = 0; k < 8; ++k)
            __builtin_nontemporal_store(d[k], dst + (size_t)k * S_DIM + j);
    }
}

extern "C" void kernel_launch(void* const* d_in, const int* in_sizes, int n_in,
                              void* d_out, int out_size, void* d_ws, size_t ws_size,
                              hipStream_t stream) {
    (void)in_sizes; (void)n_in; (void)out_size; (void)d_ws; (void)ws_size;
    const float* scores = (const float*)d_in[0];
    // d_in[1] = offset: positions[:,None]-positions[None,:] cancels it exactly.
    float* out = (float*)d_out;

    // waves = B*H * (S/16 row tiles) * (S/(16*TILES_J_PER_WAVE) strips)
    //       = 32 * 128 * 16 = 65536; 8 waves per 256-thread block.
    const int total_waves = B_DIM * H_DIM * (S_DIM / 16) * (S_DIM / (16 * TILES_J_PER_WAVE));
    const int blocks = total_waves / 8;
    alibi_wmma_kernel<<<blocks, 256, 0, stream>>>(scores, out);
}